// SimpleDGM_57896159150318
// MI455X (gfx1250) — compile-verified
//
#include <hip/hip_runtime.h>
#include <hip/hip_bf16.h>

#define N_NODES 16384
#define IN_DIM  256
#define HDIM    256
#define KNN     16

typedef float v2f __attribute__((ext_vector_type(2)));
typedef float v8f __attribute__((ext_vector_type(8)));

#define EPI_RELU     1
#define EPI_BIAS     2
#define EPI_SCALEROW 4

// ---------------------------------------------------------------------------
// Generic GEMM: C[M][N] = epilogue( A[M][K] * B[N][K]^T )
// One wave per 32x32 output block: 2x2 tiles of V_WMMA_F32_16X16X4_F32.
// Per 16-wide K chunk: 8 A-frag b64 loads + 8 B-frag b64 loads -> 16 WMMAs.
// A-frag (16x4 f32): lanes 0-15 hold K={4j,4j+1}, lanes 16-31 hold K={4j+2,4j+3}
// B-frag mirrored (N indexed by lane&15). Out-of-range B rows are clamped
// (their products land in columns that are never stored) -> no exec churn.
// ---------------------------------------------------------------------------
__global__ void gemm32_wmma_kernel(const float* __restrict__ A,
                                   const float* __restrict__ B,
                                   const float* __restrict__ bias,
                                   const float* __restrict__ rowscale,
                                   float* __restrict__ C,
                                   int M, int N, int K, int flags)
{
    int gwave = (int)((blockIdx.x * blockDim.x + threadIdx.x) >> 5);
    int lane  = (int)(threadIdx.x & 31);
    int tilesN = (N + 31) >> 5;
    int ty = gwave / tilesN;
    int tx = gwave - ty * tilesN;
    if (ty * 32 >= M) return;
    int i0 = ty * 32, n0 = tx * 32;
    int half = lane >> 4;
    int l16  = lane & 15;

    const float* Arow0 = A + (size_t)(i0 + l16)      * (size_t)K;
    const float* Arow1 = A + (size_t)(i0 + 16 + l16) * (size_t)K;
    int bn0 = n0 + l16;      if (bn0 > N - 1) bn0 = N - 1;
    int bn1 = n0 + 16 + l16; if (bn1 > N - 1) bn1 = N - 1;
    const float* Brow0 = B + (size_t)bn0 * (size_t)K;
    const float* Brow1 = B + (size_t)bn1 * (size_t)K;

    v8f acc00 = {}, acc01 = {}, acc10 = {}, acc11 = {};

    for (int kb = 0; kb < K; kb += 16) {
        v2f a0[4], a1[4], b0[4], b1[4];
#pragma unroll
        for (int j = 0; j < 4; ++j) {
            int ka = kb + 4 * j + 2 * half;
            a0[j] = *(const v2f*)(Arow0 + ka);
            a1[j] = *(const v2f*)(Arow1 + ka);
            b0[j] = *(const v2f*)(Brow0 + ka);
            b1[j] = *(const v2f*)(Brow1 + ka);
        }
#pragma unroll
        for (int j = 0; j < 4; ++j) {
            acc00 = __builtin_amdgcn_wmma_f32_16x16x4_f32(false, a0[j], false, b0[j], (short)0, acc00, false, false);
            acc01 = __builtin_amdgcn_wmma_f32_16x16x4_f32(false, a0[j], false, b1[j], (short)0, acc01, false, false);
            acc10 = __builtin_amdgcn_wmma_f32_16x16x4_f32(false, a1[j], false, b0[j], (short)0, acc10, false, false);
            acc11 = __builtin_amdgcn_wmma_f32_16x16x4_f32(false, a1[j], false, b1[j], (short)0, acc11, false, false);
        }
    }

    v8f* accs[2][2] = { { &acc00, &acc01 }, { &acc10, &acc11 } };
#pragma unroll
    for (int ni = 0; ni < 2; ++ni) {
        int n = n0 + ni * 16 + l16;
        if (n >= N) continue;
        float bv = (flags & EPI_BIAS) ? bias[n] : 0.0f;
#pragma unroll
        for (int mi = 0; mi < 2; ++mi) {
            int mbase = i0 + mi * 16 + 8 * half;
            if (mbase >= M) continue;
            v8f acc = *accs[mi][ni];
#pragma unroll
            for (int g = 0; g < 8; ++g) {
                int m = mbase + g;
                float v = acc[g];
                if (flags & EPI_SCALEROW) v *= rowscale[m];
                if (flags & EPI_BIAS)     v += bv;
                if (flags & EPI_RELU)     v = fmaxf(v, 0.0f);
                C[(size_t)m * (size_t)N + n] = v;
            }
        }
    }
}

// ---------------------------------------------------------------------------
// Row squared norms
// ---------------------------------------------------------------------------
__global__ void rownorm_kernel(const float* __restrict__ h, float* __restrict__ sq)
{
    int i = (int)(blockIdx.x * blockDim.x + threadIdx.x);
    if (i >= N_NODES) return;
    const float4* r = (const float4*)(h + (size_t)i * HDIM);
    float s = 0.0f;
#pragma unroll 4
    for (int t = 0; t < HDIM / 4; ++t) {
        float4 v = r[t];
        s += v.x * v.x + v.y * v.y + v.z * v.z + v.w * v.w;
    }
    sq[i] = s;
}

// ---------------------------------------------------------------------------
// kNN: one wave per 16-row block. Per iteration compute a 16x32 Gram slab
// (A staged in LDS, reused across both column tiles; 8 WMMAs / 12 loads).
// d2 slab -> LDS; lane r handles row r cols 0-15, lane r+16 handles row r
// cols 16-31, each keeping a private top-16 replace-max list; merged via LDS.
// ---------------------------------------------------------------------------
__global__ __launch_bounds__(32) void knn_kernel(const float* __restrict__ h,
                                                 const float* __restrict__ sq,
                                                 int* __restrict__ nbr)
{
    __shared__ float lds_h[16 * HDIM];
    __shared__ float lds_sqi[16];
    __shared__ float lds_tile[16][33];
    __shared__ float merge_d[16][KNN];
    __shared__ int   merge_i[16][KNN];

    int i0   = (int)blockIdx.x * 16;
    int lane = (int)threadIdx.x;
    int half = lane >> 4;
    int l16  = lane & 15;

    {
        const float4* src = (const float4*)(h + (size_t)i0 * HDIM);
        float4* dst = (float4*)lds_h;
        for (int t = lane; t < 16 * HDIM / 4; t += 32) dst[t] = src[t];
    }
    if (lane < 16) lds_sqi[lane] = sq[i0 + lane];
    __syncthreads();

    float sqi_r[8];
#pragma unroll
    for (int g = 0; g < 8; ++g) sqi_r[g] = lds_sqi[g + 8 * half];

    float bd[KNN]; int bi[KNN];
#pragma unroll
    for (int t = 0; t < KNN; ++t) { bd[t] = 3.4e38f; bi[t] = 0; }
    float worst = 3.4e38f; int wslot = 0;
    int myrow = i0 + l16;    // row owned by this lane (both halves)

    const float* Arow = lds_h + (size_t)l16 * HDIM;

    for (int jt = 0; jt < N_NODES / 32; ++jt) {
        int j0 = jt * 32;
        float sqj0 = sq[j0 + l16];
        float sqj1 = sq[j0 + 16 + l16];
        const float* Brow0 = h + (size_t)(j0 + l16)      * HDIM;
        const float* Brow1 = h + (size_t)(j0 + 16 + l16) * HDIM;

        // prefetch next slab's B rows (global_prefetch_b8)
        {
            int jn = j0 + 32 + lane;
            if (jn > N_NODES - 1) jn = N_NODES - 1;
            __builtin_prefetch(h + (size_t)jn * HDIM, 0, 1);
        }

        v8f acc0 = {}, acc1 = {};
        for (int kb = 0; kb < HDIM; kb += 16) {
            v2f a[4], b0[4], b1[4];
#pragma unroll
            for (int j = 0; j < 4; ++j) {
                int ka = kb + 4 * j + 2 * half;
                a[j]  = *(const v2f*)(Arow + ka);
                b0[j] = *(const v2f*)(Brow0 + ka);
                b1[j] = *(const v2f*)(Brow1 + ka);
            }
#pragma unroll
            for (int j = 0; j < 4; ++j) {
                acc0 = __builtin_amdgcn_wmma_f32_16x16x4_f32(false, a[j], false, b0[j], (short)0, acc0, false, false);
                acc1 = __builtin_amdgcn_wmma_f32_16x16x4_f32(false, a[j], false, b1[j], (short)0, acc1, false, false);
            }
        }

        __syncthreads();   // slab consumed by previous iteration
#pragma unroll
        for (int g = 0; g < 8; ++g) {
            int m = g + 8 * half;
            lds_tile[m][l16]      = sqi_r[g] + sqj0 - 2.0f * acc0[g];
            lds_tile[m][16 + l16] = sqi_r[g] + sqj1 - 2.0f * acc1[g];
        }
        __syncthreads();

        // lane (half,l16): row l16, columns half*16 .. half*16+15
#pragma unroll 1
        for (int n = 0; n < 16; ++n) {
            float d = lds_tile[l16][half * 16 + n];
            int j = j0 + half * 16 + n;
            if (j != myrow && d < worst) {
#pragma unroll
                for (int t = 0; t < KNN; ++t)
                    if (t == wslot) { bd[t] = d; bi[t] = j; }
                worst = bd[0]; wslot = 0;
#pragma unroll
                for (int t = 1; t < KNN; ++t)
                    if (bd[t] > worst) { worst = bd[t]; wslot = t; }
            }
        }
    }

    // merge the two half-lists per row: lane r+16 publishes, lane r absorbs
    __syncthreads();
    if (lane >= 16) {
#pragma unroll
        for (int t = 0; t < KNN; ++t) { merge_d[l16][t] = bd[t]; merge_i[l16][t] = bi[t]; }
    }
    __syncthreads();
    if (lane < 16) {
#pragma unroll 1
        for (int n = 0; n < KNN; ++n) {
            float d = merge_d[l16][n];
            int j = merge_i[l16][n];
            if (d < worst) {
#pragma unroll
                for (int t = 0; t < KNN; ++t)
                    if (t == wslot) { bd[t] = d; bi[t] = j; }
                worst = bd[0]; wslot = 0;
#pragma unroll
                for (int t = 1; t < KNN; ++t)
                    if (bd[t] > worst) { worst = bd[t]; wslot = t; }
            }
        }
#pragma unroll
        for (int t = 0; t < KNN; ++t) nbr[(size_t)myrow * KNN + t] = bi[t];
    }
}

// ---------------------------------------------------------------------------
// Degree / norm:  deg[i] = 1 (self) + K (own out-edges mirrored) + indeg[i]
// ---------------------------------------------------------------------------
__global__ void deg_init_kernel(int* __restrict__ deg)
{
    int i = (int)(blockIdx.x * blockDim.x + threadIdx.x);
    if (i < N_NODES) deg[i] = 1 + KNN;
}

__global__ void deg_scatter_kernel(const int* __restrict__ nbr, int* __restrict__ deg)
{
    int e = (int)(blockIdx.x * blockDim.x + threadIdx.x);
    if (e < N_NODES * KNN) atomicAdd(&deg[nbr[e]], 1);
}

__global__ void dinv_kernel(const int* __restrict__ deg, float* __restrict__ dinv)
{
    int i = (int)(blockIdx.x * blockDim.x + threadIdx.x);
    if (i < N_NODES) dinv[i] = rsqrtf((float)deg[i]);
}

// ---------------------------------------------------------------------------
// GCN aggregation. md[i][c] = dinv[i] * (hf @ W^T)[i][c]  (GEMM epilogue)
// agg[i] = md[i] + sum_{j in nbr(i)} md[j]     (gather, deterministic)
// agg[j] += md[i]  for j in nbr(i)             (scatter, fp32 atomics)
// hf'[i] = relu(dinv[i]*agg[i] + b)
// ---------------------------------------------------------------------------
__global__ void gcn_gather_kernel(const float* __restrict__ md,
                                  const int* __restrict__ nbr,
                                  float* __restrict__ agg)
{
    int wave = (int)((blockIdx.x * blockDim.x + threadIdx.x) >> 5);
    int lane = (int)(threadIdx.x & 31);
    if (wave >= N_NODES) return;
    int i = wave;
    int c = lane * 8;
    const float* base = md + (size_t)i * HDIM + c;
    float s[8];
#pragma unroll
    for (int u = 0; u < 8; ++u) s[u] = base[u];
#pragma unroll 1
    for (int t = 0; t < KNN; ++t) {
        int j = nbr[(size_t)i * KNN + t];
        const float* p = md + (size_t)j * HDIM + c;
#pragma unroll
        for (int u = 0; u < 8; ++u) s[u] += p[u];
    }
    float* o = agg + (size_t)i * HDIM + c;
#pragma unroll
    for (int u = 0; u < 8; ++u) o[u] = s[u];
}

__global__ void gcn_scatter_kernel(const float* __restrict__ md,
                                   const int* __restrict__ nbr,
                                   float* __restrict__ agg)
{
    int wave = (int)((blockIdx.x * blockDim.x + threadIdx.x) >> 5);
    int lane = (int)(threadIdx.x & 31);
    if (wave >= N_NODES) return;
    int i = wave;
    int c = lane * 8;
    const float* base = md + (size_t)i * HDIM + c;
    float v[8];
#pragma unroll
    for (int u = 0; u < 8; ++u) v[u] = base[u];
#pragma unroll 1
    for (int t = 0; t < KNN; ++t) {
        int j = nbr[(size_t)i * KNN + t];
        float* dst = agg + (size_t)j * HDIM + c;
#pragma unroll
        for (int u = 0; u < 8; ++u) atomicAdd(dst + u, v[u]);
    }
}

__global__ void gcn_finalize_kernel(const float* __restrict__ agg,
                                    const float* __restrict__ dinv,
                                    const float* __restrict__ bias,
                                    float* __restrict__ out)
{
    int t = (int)(blockIdx.x * blockDim.x + threadIdx.x);
    if (t >= N_NODES * (HDIM / 4)) return;
    int i  = t >> 6;
    int c4 = (t & 63) * 4;
    float d = dinv[i];
    const float* a = agg + (size_t)i * HDIM + c4;
    float* o = out + (size_t)i * HDIM + c4;
#pragma unroll
    for (int u = 0; u < 4; ++u)
        o[u] = fmaxf(d * a[u] + bias[c4 + u], 0.0f);
}

// ---------------------------------------------------------------------------
// Host-side orchestration
// ---------------------------------------------------------------------------
static inline size_t align256(size_t x) { return (x + 255) & ~(size_t)255; }

extern "C" void kernel_launch(void* const* d_in, const int* in_sizes, int n_in,
                              void* d_out, int out_size, void* d_ws, size_t ws_size,
                              hipStream_t stream)
{
    const float* x      = (const float*)d_in[0];   // 16384 x 256
    const float* W_pre  = (const float*)d_in[1];   // 256 x 256
    const float* b_pre  = (const float*)d_in[2];   // 256
    const float* W_conv = (const float*)d_in[3];   // 2 x 256 x 256
    const float* b_conv = (const float*)d_in[4];   // 2 x 256
    const float* W_fc1  = (const float*)d_in[5];   // 128 x 256
    const float* b_fc1  = (const float*)d_in[6];   // 128
    const float* W_fc2  = (const float*)d_in[7];   // 10 x 128
    const float* b_fc2  = (const float*)d_in[8];   // 10
    float* pred = (float*)d_out;                   // 16384 x 10

    char* ws = (char*)d_ws;
    size_t off = 0;
    float* h    = (float*)(ws + off); off = align256(off + (size_t)N_NODES * HDIM * 4);
    float* sq   = (float*)(ws + off); off = align256(off + (size_t)N_NODES * 4);
    int*   nbr  = (int*  )(ws + off); off = align256(off + (size_t)N_NODES * KNN * 4);
    int*   deg  = (int*  )(ws + off); off = align256(off + (size_t)N_NODES * 4);
    float* dinv = (float*)(ws + off); off = align256(off + (size_t)N_NODES * 4);
    float* md   = (float*)(ws + off); off = align256(off + (size_t)N_NODES * HDIM * 4);
    float* agg  = (float*)(ws + off); off = align256(off + (size_t)N_NODES * HDIM * 4);
    float* hf   = (float*)(ws + off); off = align256(off + (size_t)N_NODES * HDIM * 4);
    float* hf2  = h;     // h is dead after layer-0 GEMM; reuse
    float* hid  = md;    // md is dead after layer-1 finalize; reuse

    const int TB = 256;  // 8 waves per block
    auto gemm_blocks = [](int M, int N) {
        long long waves = (long long)((M + 31) / 32) * ((N + 31) / 32);
        return (int)((waves + 7) / 8);
    };

    // 1) h = relu(x @ W_pre^T + b_pre)
    gemm32_wmma_kernel<<<gemm_blocks(N_NODES, HDIM), TB, 0, stream>>>(
        x, W_pre, b_pre, nullptr, h, N_NODES, HDIM, IN_DIM, EPI_BIAS | EPI_RELU);

    // 2) row norms
    rownorm_kernel<<<(N_NODES + 255) / 256, 256, 0, stream>>>(h, sq);

    // 3) kNN
    knn_kernel<<<N_NODES / 16, 32, 0, stream>>>(h, sq, nbr);

    // 4) degrees / norms
    deg_init_kernel<<<(N_NODES + 255) / 256, 256, 0, stream>>>(deg);
    deg_scatter_kernel<<<(N_NODES * KNN + 255) / 256, 256, 0, stream>>>(nbr, deg);
    dinv_kernel<<<(N_NODES + 255) / 256, 256, 0, stream>>>(deg, dinv);

    // 5) two GCN layers
    const float* layer_in = h;
    float* layer_out = hf;
    for (int l = 0; l < 2; ++l) {
        const float* Wl = W_conv + (size_t)l * HDIM * HDIM;
        const float* bl = b_conv + (size_t)l * HDIM;
        gemm32_wmma_kernel<<<gemm_blocks(N_NODES, HDIM), TB, 0, stream>>>(
            layer_in, Wl, nullptr, dinv, md, N_NODES, HDIM, HDIM, EPI_SCALEROW);
        gcn_gather_kernel<<<(N_NODES * 32 + TB - 1) / TB, TB, 0, stream>>>(md, nbr, agg);
        gcn_scatter_kernel<<<(N_NODES * 32 + TB - 1) / TB, TB, 0, stream>>>(md, nbr, agg);
        gcn_finalize_kernel<<<(N_NODES * 64 + 255) / 256, 256, 0, stream>>>(agg, dinv, bl, layer_out);
        layer_in = layer_out;
        layer_out = hf2;
    }
    const float* gout = hf2;   // output of layer 1

    // 6) hid = relu(gout @ W_fc1^T + b_fc1)   [16384 x 128]
    gemm32_wmma_kernel<<<gemm_blocks(N_NODES, HDIM / 2), TB, 0, stream>>>(
        gout, W_fc1, b_fc1, nullptr, hid, N_NODES, HDIM / 2, HDIM, EPI_BIAS | EPI_RELU);

    // 7) pred = hid @ W_fc2^T + b_fc2   [16384 x 10]
    gemm32_wmma_kernel<<<gemm_blocks(N_NODES, 10), TB, 0, stream>>>(
        hid, W_fc2, b_fc2, nullptr, pred, N_NODES, 10, HDIM / 2, EPI_BIAS);

    (void)in_sizes; (void)n_in; (void)out_size; (void)ws_size;
}